// Grouper_10677288697888
// MI455X (gfx1250) — compile-verified
//
#include <hip/hip_runtime.h>
#include <stdint.h>

typedef float v2f __attribute__((ext_vector_type(2)));
typedef float v8f __attribute__((ext_vector_type(8)));
typedef unsigned int v4u __attribute__((ext_vector_type(4)));
typedef int v4i __attribute__((ext_vector_type(4)));
typedef int v8i __attribute__((ext_vector_type(8)));

constexpr int B = 8, N = 16384, G = 1024, S = 32, KORD = 4;
constexpr int BN = B * N;
constexpr int M_TOT = B * G; // 8192

// d_out flat layout (in 4-byte elements), return order of the reference tuple
constexpr int O_SIDX = 0;                              // s_idx   [8192] int32
constexpr int O_SN   = O_SIDX + M_TOT;                 // s_n     [8192,31,3] f32
constexpr int O_SXYZ = O_SN + M_TOT * (S - 1) * 3;     // s_xyz   [8192,3] f32
constexpr int O_SORD = O_SXYZ + M_TOT * 3;             // s_order [4,8192] int32
constexpr int O_SINV = O_SORD + KORD * M_TOT;          // s_inverse

// ---------------------------------------------------------------------------
// Kernel 1: farthest point sampling, one workgroup per batch.
// Per-thread state fully in registers: 16 points (x,y,z) + 16 running dists.
// LDS holds the 1024-wide argmax tree, the selected-index list and the broadcast of the last
// selected point. Sequential loop of G-1 steps (latency bound by design).
// ---------------------------------------------------------------------------
__global__ __launch_bounds__(1024)
void fps_kernel(const float* __restrict__ coord,
                int* __restrict__ s_idx_out,
                float* __restrict__ sxyz_out) {
  const int b = blockIdx.x;
  const int tid = threadIdx.x;
  const float* pts = coord + (size_t)b * N * 3;

  __shared__ float rv[1024];
  __shared__ int   ri[1024];
  __shared__ float lastp[3];
  __shared__ int   selSh[G];

  float px[16], py[16], pz[16], dloc[16];
#pragma unroll
  for (int k = 0; k < 16; ++k) {
    int i = tid + k * 1024;
    px[k] = pts[i * 3 + 0];
    py[k] = pts[i * 3 + 1];
    pz[k] = pts[i * 3 + 2];
    dloc[k] = 3.4e38f;
  }
  if (tid == 0) {
    selSh[0] = 0;
    lastp[0] = pts[0]; lastp[1] = pts[1]; lastp[2] = pts[2];
  }
  __syncthreads();

  for (int g = 1; g < G; ++g) {
    const float lx = lastp[0], ly = lastp[1], lz = lastp[2];
    float bv = -1.0f; int bi = 0x7fffffff;
#pragma unroll
    for (int k = 0; k < 16; ++k) {
      float dx = px[k] - lx, dy = py[k] - ly, dz = pz[k] - lz;
      float d = dx * dx + dy * dy + dz * dz;
      float nd = fminf(dloc[k], d);
      dloc[k] = nd;
      int i = tid + k * 1024;
      if (nd > bv) { bv = nd; bi = i; }
    }
    rv[tid] = bv; ri[tid] = bi;
    __syncthreads();
    for (int s = 512; s > 0; s >>= 1) {
      if (tid < s) {
        float ov = rv[tid + s]; int oi = ri[tid + s];
        float mv = rv[tid];     int mi = ri[tid];
        if (ov > mv || (ov == mv && oi < mi)) { rv[tid] = ov; ri[tid] = oi; }
      }
      __syncthreads();
    }
    if (tid == 0) {
      int sel = ri[0];
      selSh[g] = sel;
      lastp[0] = pts[sel * 3 + 0];
      lastp[1] = pts[sel * 3 + 1];
      lastp[2] = pts[sel * 3 + 2];
    }
    __syncthreads();
  }

  // emit s_idx (global index) and s_xyz
  {
    int i = tid;              // G == blockDim.x == 1024
    int sel = selSh[i];
    s_idx_out[b * G + i] = b * N + sel;
    sxyz_out[(b * G + i) * 3 + 0] = pts[sel * 3 + 0];
    sxyz_out[(b * G + i) * 3 + 1] = pts[sel * 3 + 1];
    sxyz_out[(b * G + i) * 3 + 2] = pts[sel * 3 + 2];
  }
}

// ---------------------------------------------------------------------------
// TDM helper: stage nElem f32 elements from global into LDS (1-D tile).
// D# per CDNA5 ISA:
//   group0: [d0]=count=1, [d1]=lds byte addr, [d2]=gaddr lo,
//           [d3]=gaddr[56:32] | type=2<<30
//   group1: [d0]=data_size(4B)<<16, tensor_dim0=tile_dim0=nElem,
//           tensor_dim1=1, tile_dim1=1, tensor_dim*_stride=nElem
//   remaining groups: zero (1-D tile, dims 2..4 unused)
// Tracked with TENSORcnt; EXEC is ignored by TDM ops (issued per wave).
// This toolchain exposes the 6-arg builtin:
//   (uint32x4, int32x8, int32x4, int32x4, int32x8, i32 cpol)
// ---------------------------------------------------------------------------
__device__ __forceinline__ void tdm_load_f32(const float* gsrc,
                                             unsigned ldsByteAddr,
                                             unsigned nElem) {
  unsigned long long ga = (unsigned long long)(const void*)gsrc;
  v4u g0;
  g0[0] = 1u;                                         // count=1, user mode
  g0[1] = ldsByteAddr;                                // lds_addr
  g0[2] = (unsigned)ga;                               // global_addr[31:0]
  g0[3] = (unsigned)((ga >> 32) & 0x01FFFFFFu) | (2u << 30); // [56:32] | type=2
  v8i g1;
  g1[0] = (int)(2u << 16);                            // data_size = 4 bytes
  g1[1] = (int)((nElem & 0xFFFFu) << 16);             // tensor_dim0[15:0]
  g1[2] = (int)(((nElem >> 16) & 0xFFFFu) | (1u << 16)); // dim0 hi | tensor_dim1=1
  g1[3] = (int)(nElem << 16);                         // tile_dim0
  g1[4] = 1;                                          // tile_dim1 = 1
  g1[5] = (int)nElem;                                 // tensor_dim0_stride lo
  g1[6] = (int)((nElem & 0xFFFFu) << 16);             // tensor_dim1_stride[15:0]
  g1[7] = 0;
  v4i z4 = {0, 0, 0, 0};
  v8i z8 = {0, 0, 0, 0, 0, 0, 0, 0};
  __builtin_amdgcn_tensor_load_to_lds(g0, g1, z4, z4, z8, 0);
}

// ---------------------------------------------------------------------------
// Kernel 2: KNN via V_WMMA_F32_16X16X4_F32, fed by TDM double buffering.
// Block = 8 waves, one batch; waves cooperatively stage 1024 points (12 KB)
// per iteration into a 2-buffer LDS pipeline with tensor_load_to_lds
// (each wave moves its 128-point slice; s_wait_tensorcnt + barrier gate
// consumption). Each wave owns 16 centers: A-tile = centers 16x4 (x,y,z,0),
// B-tile = 16 points from LDS, WMMA produces the 16x16 dot tile; d2 is
// finished with |c|^2+|p|^2-2*dot. Lanes 0-15 keep a register-resident
// top-32 (dist,idx) list per center (all indices static / unrolled).
// ---------------------------------------------------------------------------
__global__ __launch_bounds__(256)
void knn_kernel(const float* __restrict__ coord,
                const float* __restrict__ sxyz,
                int* __restrict__ nidx) {
  const int tid   = threadIdx.x;
  const int wave  = tid >> 5;
  const int lane  = tid & 31;
  const int m     = lane & 15;
  const int upper = lane >> 4;

  const int batch = blockIdx.x >> 3;                 // 8 blocks per batch
  const int tile  = (blockIdx.x & 7) * 8 + wave;     // 64 center-tiles per batch
  const int cbase = batch * G + tile * 16;           // global center row base

  constexpr int MC  = 1024;        // points staged per pipeline stage
  constexpr int NMC = N / MC;      // 16 stages
  constexpr int PPW = MC / 8;      // 128 points per wave per stage

  __shared__ float ptsSh[2][MC * 3];                 // 24 KB double buffer
  __shared__ float tileSh[8][16][17];                // +1 pad: no bank conflicts
  __shared__ float cnSh[8][16];

  // Centers; WMMA A-tile (16x4 f32): lanes 0-15: VGPR0=K0(x), VGPR1=K1(y);
  // lanes 16-31: VGPR0=K2(z), VGPR1=K3(0).
  float cx = sxyz[(cbase + m) * 3 + 0];
  float cy = sxyz[(cbase + m) * 3 + 1];
  float cz = sxyz[(cbase + m) * 3 + 2];
  if (!upper) cnSh[wave][m] = cx * cx + cy * cy + cz * cz;
  asm volatile("s_wait_dscnt 0" ::: "memory");
  float cnr[8];
#pragma unroll
  for (int r = 0; r < 8; ++r) cnr[r] = cnSh[wave][upper * 8 + r];

  v2f a;
  a.x = upper ? cz : cx;
  a.y = upper ? 0.0f : cy;

  // top-32 candidate list (registers only; all index expressions static)
  float bd[S]; int bix[S];
#pragma unroll
  for (int k = 0; k < S; ++k) { bd[k] = 3.4e38f; bix[k] = 0; }
  float wmax = 3.4e38f; int wpos = 0;

  const float* pts = coord + (size_t)batch * N * 3;

  // stage(mc, buf): this wave DMAs its 128-point slice of mega-chunk mc
  auto stage = [&](int mc, int buf) {
    const float* src = pts + (size_t)(mc * MC + wave * PPW) * 3;
    unsigned ldsAddr = (unsigned)(size_t)(const void*)&ptsSh[buf][wave * PPW * 3];
    tdm_load_f32(src, ldsAddr, PPW * 3);
  };

  stage(0, 0);
  for (int mc = 0; mc < NMC; ++mc) {
    const int cur = mc & 1;
    if (mc + 1 < NMC) {
      stage(mc + 1, cur ^ 1);                 // kick next stage, then
      __builtin_amdgcn_s_wait_tensorcnt(1);   // wait for current stage only
    } else {
      __builtin_amdgcn_s_wait_tensorcnt(0);
    }
    __syncthreads();                          // all waves' slices resident

    const float* P = &ptsSh[cur][0];
    for (int sub = 0; sub < MC / 16; ++sub) {
      const int p = sub * 16 + m;
      float qx = P[p * 3 + 0];
      float qy = P[p * 3 + 1];
      float qz = P[p * 3 + 2];
      float pn = qx * qx + qy * qy + qz * qz;
      // B-tile (4x16 f32): lanes 0-15: K0(x),K1(y); lanes 16-31: K2(z),K3(0)
      v2f bb;
      bb.x = upper ? qz : qx;
      bb.y = upper ? 0.0f : qy;

      v8f acc = {};
      acc = __builtin_amdgcn_wmma_f32_16x16x4_f32(
          /*neg_a=*/false, a, /*neg_b=*/false, bb,
          /*c_mod=*/(short)0, acc, /*reuse_a=*/false, /*reuse_b=*/false);

      // C/D layout: VGPR r, lanes 0-15 -> M=r, lanes 16-31 -> M=8+r; N=lane&15
#pragma unroll
      for (int r = 0; r < 8; ++r) {
        int mm = upper * 8 + r;
        float d2 = cnr[r] + pn - 2.0f * acc[r];
        tileSh[wave][mm][m] = d2;
      }
      asm volatile("s_wait_dscnt 0" ::: "memory");  // intra-wave tile handoff

      if (!upper) {
        const int n0 = mc * MC + sub * 16;
#pragma unroll
        for (int j = 0; j < 16; ++j) {
          float d = tileSh[wave][m][j];
          if (d < wmax) {
            int idx = n0 + j;
#pragma unroll
            for (int k = 0; k < S; ++k) {
              bd[k]  = (k == wpos) ? d   : bd[k];
              bix[k] = (k == wpos) ? idx : bix[k];
            }
            wmax = -1.0f; wpos = 0;   // rescan for new running max
#pragma unroll
            for (int k = 0; k < S; ++k) {
              if (bd[k] > wmax) { wmax = bd[k]; wpos = k; }
            }
          }
        }
      }
    }
    __syncthreads();   // everyone done with buffer `cur` before it is refilled
  }

  if (!upper) {
    // sort the 32 candidates ascending by (d, idx) — unrolled bitonic network
#pragma unroll
    for (int k = 2; k <= S; k <<= 1) {
#pragma unroll
      for (int j = k >> 1; j > 0; j >>= 1) {
#pragma unroll
        for (int i = 0; i < S; ++i) {
          int l = i ^ j;
          if (l > i) {
            bool up = ((i & k) == 0);
            bool gt = (bd[i] > bd[l]) || (bd[i] == bd[l] && bix[i] > bix[l]);
            if (up == gt) {
              float td = bd[i]; bd[i] = bd[l]; bd[l] = td;
              int   ti = bix[i]; bix[i] = bix[l]; bix[l] = ti;
            }
          }
        }
      }
    }
    const int g = cbase + m;
#pragma unroll
    for (int j = 0; j < S; ++j) nidx[g * S + j] = batch * N + bix[j];
  }
}

// ---------------------------------------------------------------------------
// Kernel 3: s_n = coord[nbr] - center, dropping neighbor 0 (the center itself)
// ---------------------------------------------------------------------------
__global__ void gather_sn(const float* __restrict__ coord,
                          const float* __restrict__ sxyz,
                          const int* __restrict__ nidx,
                          float* __restrict__ sn) {
  int t = blockIdx.x * blockDim.x + threadIdx.x;
  const int total = M_TOT * (S - 1);
  if (t >= total) return;
  int g = t / (S - 1);
  int j = t - g * (S - 1) + 1;
  int p = nidx[g * S + j];
#pragma unroll
  for (int c = 0; c < 3; ++c)
    sn[(size_t)t * 3 + c] = coord[(size_t)p * 3 + c] - sxyz[g * 3 + c];
}

// ---------------------------------------------------------------------------
// Kernel 4: stable argsort of code_sel rows (4 x 8192) + inverse permutation.
// One workgroup per row; LDS-resident bitonic sort of (code<<13 | idx) u64
// keys (64 KB LDS — comfortable in the 320 KB WGP LDS).
// ---------------------------------------------------------------------------
__global__ __launch_bounds__(1024)
void order_kernel(const int* __restrict__ code,
                  const int* __restrict__ s_idx,
                  int* __restrict__ s_order,
                  int* __restrict__ s_inverse) {
  const int row = blockIdx.x;
  const int tid = threadIdx.x;
  __shared__ unsigned long long key[M_TOT];

  for (int i = tid; i < M_TOT; i += 1024) {
    int sg = s_idx[i];
    unsigned int k = (unsigned int)code[(size_t)row * BN + sg];  // < 2^30
    key[i] = ((unsigned long long)k << 13) | (unsigned long long)i;
  }
  __syncthreads();

  for (int ksz = 2; ksz <= M_TOT; ksz <<= 1) {
    for (int j = ksz >> 1; j > 0; j >>= 1) {
      for (int i = tid; i < M_TOT; i += 1024) {
        int l = i ^ j;
        if (l > i) {
          bool up = ((i & ksz) == 0);
          unsigned long long a = key[i], c = key[l];
          if (up == (a > c)) { key[i] = c; key[l] = a; }
        }
      }
      __syncthreads();
    }
  }

  for (int i = tid; i < M_TOT; i += 1024) {
    int idx = (int)(key[i] & 8191ull);
    s_order[row * M_TOT + i] = idx;
    s_inverse[row * M_TOT + idx] = i;
  }
}

// ---------------------------------------------------------------------------
extern "C" void kernel_launch(void* const* d_in, const int* in_sizes, int n_in,
                              void* d_out, int out_size, void* d_ws, size_t ws_size,
                              hipStream_t stream) {
  const float* coord = (const float*)d_in[0];
  const int*   code  = (const int*)d_in[1];
  (void)in_sizes; (void)n_in; (void)out_size; (void)ws_size;

  float* outf     = (float*)d_out;
  int*   sidx_out = (int*)d_out + O_SIDX;
  float* sn_out   = outf + O_SN;
  float* sxyz_out = outf + O_SXYZ;
  int*   sord     = (int*)d_out + O_SORD;
  int*   sinv     = (int*)d_out + O_SINV;
  int*   nidx     = (int*)d_ws;   // [8192][32] neighbor indices (1 MB)

  fps_kernel<<<B, 1024, 0, stream>>>(coord, sidx_out, sxyz_out);
  knn_kernel<<<64, 256, 0, stream>>>(coord, sxyz_out, nidx);

  const int total = M_TOT * (S - 1);
  gather_sn<<<(total + 255) / 256, 256, 0, stream>>>(coord, sxyz_out, nidx, sn_out);
  order_kernel<<<KORD, 1024, 0, stream>>>(code, sidx_out, sord, sinv);
}